// HierEncoder_65721589563554
// MI455X (gfx1250) — compile-verified
//
#include <hip/hip_runtime.h>

typedef __attribute__((ext_vector_type(2))) float v2f;
typedef __attribute__((ext_vector_type(4))) float v4f;
typedef __attribute__((ext_vector_type(8))) float v8f;

#define N_SOURCE   100000
#define N_TARGET   4096
#define DEG        800
#define SOURCE_DIM 512
#define EMBED_DIM  32
#define TARGET_DIM 16

// ---------------------------------------------------------------------------
// Kernel 0: transpose embed [512,32] -> embedT [32,512] so B-fragment loads
// in the big GEMM are contiguous b128 per lane.
// ---------------------------------------------------------------------------
__global__ void __launch_bounds__(256)
transpose_embed(const float* __restrict__ embed, float* __restrict__ embedT) {
    int idx = blockIdx.x * blockDim.x + threadIdx.x;
    if (idx < SOURCE_DIM * EMBED_DIM) {
        int k = idx >> 5;   // row in embed (0..511)
        int n = idx & 31;   // col in embed (0..31)
        embedT[n * SOURCE_DIM + k] = embed[idx];
    }
}

// ---------------------------------------------------------------------------
// Kernel 1: X = (source_feat @ embed) / x_norm[:,None]   via fp32 WMMA.
//
// One wave handles TWO 16-row M tiles (32 rows) x both 16-wide N tiles:
// 4 accumulators, B fragments reused across the M tiles.
//
// K-permutation trick: the 16x4 A fragment wants lane-half hi to hold
// K = 4kb+2hi, +1 — not contiguous across steps. Since the K reduction
// order is free, we re-map K per 8-block:
//   even step: hi=0 -> K=8k+0,1   hi=1 -> K=8k+4,5
//   odd  step: hi=0 -> K=8k+2,3   hi=1 -> K=8k+6,7
// so ONE global_load_b128 at (8k + 4*hi) feeds two consecutive WMMA steps,
// for A and (identically permuted) BT. Halves VMEM instruction count.
// ---------------------------------------------------------------------------
__global__ void __launch_bounds__(256)
embed_gemm(const float* __restrict__ A, const float* __restrict__ BT,
           const float* __restrict__ xnorm, float* __restrict__ X) {
    const int wave = threadIdx.x >> 5;
    const int tile = blockIdx.x * 8 + wave;          // 32-row tile index
    if (tile >= N_SOURCE / 32) return;               // wave-uniform guard
    const int lane = threadIdx.x & 31;
    const int lh   = lane & 15;
    const int hi   = lane >> 4;                      // 0 or 1
    const int row0 = tile * 32;

    const float* __restrict__ arow0 = A  + (size_t)(row0 + lh)      * SOURCE_DIM + 4 * hi;
    const float* __restrict__ arow1 = A  + (size_t)(row0 + 16 + lh) * SOURCE_DIM + 4 * hi;
    const float* __restrict__ b0row = BT + (size_t)(lh)             * SOURCE_DIM + 4 * hi;
    const float* __restrict__ b1row = BT + (size_t)(16 + lh)        * SOURCE_DIM + 4 * hi;

    v8f c00 = {}, c01 = {}, c10 = {}, c11 = {};

    for (int k = 0; k < SOURCE_DIM / 8; ++k) {
        v4f a0 = *(const v4f*)(arow0 + 8 * k);
        v4f a1 = *(const v4f*)(arow1 + 8 * k);
        v4f b0 = *(const v4f*)(b0row + 8 * k);
        v4f b1 = *(const v4f*)(b1row + 8 * k);
        // unconditional speculative prefetch ~1KB ahead of the A streams
        __builtin_prefetch(arow0 + 8 * k + 256, 0, 3);
        __builtin_prefetch(arow1 + 8 * k + 256, 0, 3);

        v2f a0e = {a0.x, a0.y}, a0o = {a0.z, a0.w};
        v2f a1e = {a1.x, a1.y}, a1o = {a1.z, a1.w};
        v2f b0e = {b0.x, b0.y}, b0o = {b0.z, b0.w};
        v2f b1e = {b1.x, b1.y}, b1o = {b1.z, b1.w};

        c00 = __builtin_amdgcn_wmma_f32_16x16x4_f32(false, a0e, false, b0e,
                                                    (short)0, c00, false, false);
        c01 = __builtin_amdgcn_wmma_f32_16x16x4_f32(false, a0e, false, b1e,
                                                    (short)0, c01, false, false);
        c10 = __builtin_amdgcn_wmma_f32_16x16x4_f32(false, a1e, false, b0e,
                                                    (short)0, c10, false, false);
        c11 = __builtin_amdgcn_wmma_f32_16x16x4_f32(false, a1e, false, b1e,
                                                    (short)0, c11, false, false);
        c00 = __builtin_amdgcn_wmma_f32_16x16x4_f32(false, a0o, false, b0o,
                                                    (short)0, c00, false, false);
        c01 = __builtin_amdgcn_wmma_f32_16x16x4_f32(false, a0o, false, b1o,
                                                    (short)0, c01, false, false);
        c10 = __builtin_amdgcn_wmma_f32_16x16x4_f32(false, a1o, false, b0o,
                                                    (short)0, c10, false, false);
        c11 = __builtin_amdgcn_wmma_f32_16x16x4_f32(false, a1o, false, b1o,
                                                    (short)0, c11, false, false);
    }

    // fuse the row normalization and store X
#pragma unroll
    for (int r = 0; r < 8; ++r) {
        int   rowA = row0 + r + 8 * hi;
        int   rowB = rowA + 16;
        float invA = 1.0f / xnorm[rowA];
        float invB = 1.0f / xnorm[rowB];
        X[(size_t)rowA * EMBED_DIM + lh]      = c00[r] * invA;
        X[(size_t)rowA * EMBED_DIM + 16 + lh] = c01[r] * invA;
        X[(size_t)rowB * EMBED_DIM + lh]      = c10[r] * invB;
        X[(size_t)rowB * EMBED_DIM + 16 + lh] = c11[r] * invB;
    }
}

// ---------------------------------------------------------------------------
// Kernel 2: segment mean. One block per target; each wave reads one gathered
// 32-float row per step (fully coalesced 128B from L2-resident X); 8 waves
// stride the edge block; LDS reduction across waves; fused 1/deg.
// ---------------------------------------------------------------------------
__global__ void __launch_bounds__(256)
gather_mean(const float* __restrict__ X, const int* __restrict__ src,
            const int* __restrict__ range_list, float* __restrict__ agg) {
    __shared__ float red[8][EMBED_DIM];
    const int t     = blockIdx.x;
    const int col   = threadIdx.x & 31;
    const int grp   = threadIdx.x >> 5;
    const int start = range_list[2 * t];
    const int end   = range_list[2 * t + 1];

    float acc = 0.0f;
    for (int e = start + grp; e < end; e += 8) {
        int s = src[e];
        acc += X[(size_t)s * EMBED_DIM + col];
    }
    red[grp][col] = acc;
    __syncthreads();
    if (grp == 0) {
        float sum = acc;
#pragma unroll
        for (int g = 1; g < 8; ++g) sum += red[g][col];
        float deg = (float)(end - start);
        deg = deg > 1.0f ? deg : 1.0f;
        agg[(size_t)t * EMBED_DIM + col] = sum / deg;
    }
}

// ---------------------------------------------------------------------------
// Kernel 3: out = agg @ weight  ([4096,32]x[32,16]) via fp32 WMMA.
// One wave per 16-row tile; N=16 is exactly one WMMA tile; K=32 -> 8 steps.
// ---------------------------------------------------------------------------
__global__ void __launch_bounds__(256)
proj_gemm(const float* __restrict__ agg, const float* __restrict__ W,
          float* __restrict__ out) {
    const int wave = threadIdx.x >> 5;
    const int tile = blockIdx.x * 8 + wave;
    if (tile >= N_TARGET / 16) return;
    const int lane = threadIdx.x & 31;
    const int lh   = lane & 15;
    const int hi   = lane >> 4;
    const int row0 = tile * 16;

    const float* __restrict__ arow = agg + (size_t)(row0 + lh) * EMBED_DIM + 2 * hi;

    v8f c = {};
#pragma unroll
    for (int kb = 0; kb < EMBED_DIM / 4; ++kb) {
        v2f a = *(const v2f*)(arow + 4 * kb);
        v2f b;
        b.x = W[(4 * kb + 2 * hi) * TARGET_DIM + lh];
        b.y = W[(4 * kb + 2 * hi + 1) * TARGET_DIM + lh];
        c = __builtin_amdgcn_wmma_f32_16x16x4_f32(false, a, false, b,
                                                  (short)0, c, false, false);
    }

#pragma unroll
    for (int r = 0; r < 8; ++r) {
        int row = row0 + r + 8 * hi;
        out[(size_t)row * TARGET_DIM + lh] = c[r];
    }
}

// ---------------------------------------------------------------------------
extern "C" void kernel_launch(void* const* d_in, const int* in_sizes, int n_in,
                              void* d_out, int out_size, void* d_ws, size_t ws_size,
                              hipStream_t stream) {
    (void)in_sizes; (void)n_in; (void)out_size; (void)ws_size;

    const float* source_feat = (const float*)d_in[0];
    const float* embed       = (const float*)d_in[1];
    const float* weight      = (const float*)d_in[2];
    const int*   edge_index  = (const int*)d_in[3];   // row 0 = src, row 1 = tgt
    const int*   range_list  = (const int*)d_in[4];
    const float* x_norm      = (const float*)d_in[5];
    float*       out         = (float*)d_out;

    // workspace layout
    float* embedT = (float*)d_ws;                                          // 64 KB
    float* X      = (float*)((char*)d_ws + 65536);                         // 12.8 MB
    float* agg    = (float*)((char*)d_ws + 65536 +
                             (size_t)N_SOURCE * EMBED_DIM * sizeof(float)); // 0.5 MB

    const int* src = edge_index;  // first N_EDGES entries of edge_index

    transpose_embed<<<(SOURCE_DIM * EMBED_DIM + 255) / 256, 256, 0, stream>>>(embed, embedT);

    embed_gemm<<<(N_SOURCE / 32 + 7) / 8, 256, 0, stream>>>(source_feat, embedT, x_norm, X);

    gather_mean<<<N_TARGET, 256, 0, stream>>>(X, src, range_list, agg);

    proj_gemm<<<(N_TARGET / 16) / 8, 256, 0, stream>>>(agg, weight, out);
}